// GATNet_17695265259750
// MI455X (gfx1250) — compile-verified
//
#include <hip/hip_runtime.h>
#include <hip/hip_bf16.h>

// ---------------- problem constants (match reference) ----------------
#define NN      50000
#define LL      10
#define INC     300
#define HEADS   4
#define OUTC    64
#define HD      256          // HEADS*OUTC
#define EE      800000
#define NEG_SLOPE 0.2f

typedef float v8f __attribute__((ext_vector_type(8)));
typedef float v2f __attribute__((ext_vector_type(2)));

__device__ __forceinline__ float lrelu(float a) {
    return a > 0.0f ? a : NEG_SLOPE * a;
}
// monotone float <-> uint encoding so atomicMax(uint) == float max
__device__ __forceinline__ unsigned encf(float f) {
    unsigned u = __float_as_uint(f);
    return (u & 0x80000000u) ? ~u : (u | 0x80000000u);
}
__device__ __forceinline__ float decf(unsigned u) {
    return __uint_as_float((u & 0x80000000u) ? (u & 0x7FFFFFFFu) : ~u);
}

// K1: xm[n,c] = mean over L of x[n,l,c]   (streams 600 MB, HBM-bound)
__global__ __launch_bounds__(256) void mean_l_kernel(const float* __restrict__ x,
                                                     float* __restrict__ xm) {
    long long i = (long long)blockIdx.x * 256 + threadIdx.x;
    if (i >= (long long)NN * INC) return;
    int n = (int)(i / INC);
    int c = (int)(i % INC);
    const float* p = x + (long long)n * (LL * INC) + c;
    float s = 0.0f;
#pragma unroll
    for (int l = 0; l < LL; ++l) s += p[l * INC];
    xm[i] = s * (1.0f / (float)LL);
}

// K2: W2 = lin_w @ gat_w  [300x256], b2 = lin_b @ gat_w [256]  (tiny)
__global__ __launch_bounds__(256) void fuse_weights_kernel(const float* __restrict__ lin_w,
                                                           const float* __restrict__ lin_b,
                                                           const float* __restrict__ gat_w,
                                                           float* __restrict__ W2,
                                                           float* __restrict__ b2) {
    int i = blockIdx.x * 256 + threadIdx.x;
    if (i < INC * HD) {
        int r = i >> 8;          // /256
        int c = i & (HD - 1);
        float s = 0.0f;
        for (int k = 0; k < INC; ++k) s += lin_w[r * INC + k] * gat_w[k * HD + c];
        W2[i] = s;
    } else if (i < INC * HD + HD) {
        int c = i - INC * HD;
        float s = 0.0f;
        for (int k = 0; k < INC; ++k) s += lin_b[k] * gat_w[k * HD + c];
        b2[c] = s;
    }
}

// K3: xt[N,256] = xm[N,300] @ W2[300,256] + b2  via V_WMMA_F32_16X16X4_F32.
// One wave computes a 16x64 output strip (4 C tiles share one A fragment).
// A-frag layout (ISA 7.12.2, 32-bit 16x4): lanes 0-15 hold K={k,k+1},
// lanes 16-31 hold K={k+2,k+3}; B symmetric; C: vgpr v -> M=v / M=v+8.
#define ROWTILES (NN / 16)            // 3125
#define TOTWAVES (ROWTILES * 4)       // 12500
__global__ __launch_bounds__(256) void xt_gemm_wmma(const float* __restrict__ xm,
                                                    const float* __restrict__ W2,
                                                    const float* __restrict__ b2,
                                                    float* __restrict__ xt) {
    const int lane = threadIdx.x & 31;
    const int gw   = blockIdx.x * 8 + (threadIdx.x >> 5);
    if (gw >= TOTWAVES) return;                 // wave-uniform: EXEC stays all-ones
    const int rowTile = gw >> 2;
    const int colGrp  = gw & 3;                 // 64 columns each
    const int m     = rowTile * 16 + (lane & 15);
    const int koff  = (lane >> 4) << 1;         // 0 (lanes 0-15) or 2 (lanes 16-31)
    const int nbase = colGrp * 64 + (lane & 15);

    v8f acc0 = {}, acc1 = {}, acc2 = {}, acc3 = {};
    const float* __restrict__ xrow = xm + m * INC;

    for (int k = 0; k < INC; k += 4) {
        v2f a;
        a.x = xrow[k + koff];
        a.y = xrow[k + koff + 1];
        const float* B0 = W2 + (k + koff)     * HD + nbase;
        const float* B1 = W2 + (k + koff + 1) * HD + nbase;
        v2f b0  = { B0[0],  B1[0]  };
        v2f b1  = { B0[16], B1[16] };
        v2f b2v = { B0[32], B1[32] };
        v2f b3  = { B0[48], B1[48] };
        acc0 = __builtin_amdgcn_wmma_f32_16x16x4_f32(false, a, false, b0,  (short)0, acc0, false, false);
        acc1 = __builtin_amdgcn_wmma_f32_16x16x4_f32(false, a, false, b1,  (short)0, acc1, false, false);
        acc2 = __builtin_amdgcn_wmma_f32_16x16x4_f32(false, a, false, b2v, (short)0, acc2, false, false);
        acc3 = __builtin_amdgcn_wmma_f32_16x16x4_f32(false, a, false, b3,  (short)0, acc3, false, false);
    }
    const int mrow0 = rowTile * 16 + ((lane >> 4) << 3);   // +0 or +8
    const int c0    = colGrp * 64 + (lane & 15);
#pragma unroll
    for (int v = 0; v < 8; ++v) {
        int r = mrow0 + v;
        xt[r * HD + c0]      = acc0[v] + b2[c0];
        xt[r * HD + c0 + 16] = acc1[v] + b2[c0 + 16];
        xt[r * HD + c0 + 32] = acc2[v] + b2[c0 + 32];
        xt[r * HD + c0 + 48] = acc3[v] + b2[c0 + 48];
    }
}

// K4: per-node attention scalars.  att flat layout: att[h*128 + d] (x_i half),
// att[h*128 + 64 + d] (x_j half).  One block per node.
__global__ __launch_bounds__(256) void att_scalars_kernel(const float* __restrict__ xt,
                                                          const float* __restrict__ att,
                                                          float* __restrict__ a_i,
                                                          float* __restrict__ a_j) {
    __shared__ float si[256], sj[256];
    const int n = blockIdx.x;
    const int t = threadIdx.x;
    const int h = t >> 6, d = t & 63;
    float v = xt[n * HD + t];
    si[t] = v * att[h * 128 + d];
    sj[t] = v * att[h * 128 + 64 + d];
    __syncthreads();
    for (int off = 32; off >= 1; off >>= 1) {
        if (d < off) { si[t] += si[t + off]; sj[t] += sj[t + off]; }
        __syncthreads();
    }
    if (d == 0) { a_i[n * HEADS + h] = si[t]; a_j[n * HEADS + h] = sj[t]; }
}

// K5: self-loop logit + segment-max init
__global__ __launch_bounds__(256) void self_init_kernel(const float* __restrict__ a_i,
                                                        const float* __restrict__ a_j,
                                                        float* __restrict__ selfAlpha,
                                                        unsigned* __restrict__ segmax_bits) {
    int i = blockIdx.x * 256 + threadIdx.x;
    if (i >= NN * HEADS) return;
    float a = lrelu(a_i[i] + a_j[i]);
    selfAlpha[i] = a;
    segmax_bits[i] = encf(a);
}

// K6: edge pass A — segment max via uint atomicMax
__global__ __launch_bounds__(256) void edge_max_kernel(const int* __restrict__ ei,
                                                       const float* __restrict__ a_i,
                                                       const float* __restrict__ a_j,
                                                       unsigned* __restrict__ segmax_bits) {
    int i = blockIdx.x * 256 + threadIdx.x;
    if (i >= EE * HEADS) return;
    int e = i >> 2, h = i & 3;
    int r = ei[e], c = ei[EE + e];
    float a = lrelu(a_i[c * HEADS + h] + a_j[r * HEADS + h]);
    atomicMax(&segmax_bits[c * HEADS + h], encf(a));
}

// K7: decode max, init segment sums with self-loop term
__global__ __launch_bounds__(256) void segsum_init_kernel(const unsigned* __restrict__ segmax_bits,
                                                          const float* __restrict__ selfAlpha,
                                                          float* __restrict__ segmaxf,
                                                          float* __restrict__ segsum) {
    int i = blockIdx.x * 256 + threadIdx.x;
    if (i >= NN * HEADS) return;
    float mx = decf(segmax_bits[i]);
    segmaxf[i] = mx;
    segsum[i]  = __expf(selfAlpha[i] - mx);
}

// K8: edge pass B — accumulate exp(alpha - max) into segment sums
__global__ __launch_bounds__(256) void edge_sum_kernel(const int* __restrict__ ei,
                                                       const float* __restrict__ a_i,
                                                       const float* __restrict__ a_j,
                                                       const float* __restrict__ segmaxf,
                                                       float* __restrict__ segsum) {
    int i = blockIdx.x * 256 + threadIdx.x;
    if (i >= EE * HEADS) return;
    int e = i >> 2, h = i & 3;
    int r = ei[e], c = ei[EE + e];
    float a = lrelu(a_i[c * HEADS + h] + a_j[r * HEADS + h]);
    atomicAdd(&segsum[c * HEADS + h], __expf(a - segmaxf[c * HEADS + h]));
}

// K9: out init = gat_bias + selfWeight * xt  (covers every output element)
__global__ __launch_bounds__(256) void out_init_kernel(const float* __restrict__ selfAlpha,
                                                       const float* __restrict__ segmaxf,
                                                       const float* __restrict__ segsum,
                                                       const float* __restrict__ xt,
                                                       const float* __restrict__ gat_bias,
                                                       float* __restrict__ out) {
    long long i = (long long)blockIdx.x * 256 + threadIdx.x;
    if (i >= (long long)NN * HD) return;
    int n = (int)(i >> 8);
    int t = (int)(i & (HD - 1));
    int h = t >> 6;
    int nh = n * HEADS + h;
    float w = __expf(selfAlpha[nh] - segmaxf[nh]) / segsum[nh];
    out[i] = gat_bias[t] + w * xt[i];
}

// K10: edge pass C — weighted aggregation, one block per edge.
// coef per head computed once in LDS; 256 atomic f32 adds into L2-resident out.
__global__ __launch_bounds__(256) void edge_aggr_kernel(const int* __restrict__ ei,
                                                        const float* __restrict__ a_i,
                                                        const float* __restrict__ a_j,
                                                        const float* __restrict__ segmaxf,
                                                        const float* __restrict__ segsum,
                                                        const float* __restrict__ xt,
                                                        float* __restrict__ out) {
    __shared__ float coef[HEADS];
    __shared__ int sr, sc;
    const int e = blockIdx.x;
    const int t = threadIdx.x;
    if (t == 0) { sr = ei[e]; sc = ei[EE + e]; }
    __syncthreads();
    const int r = sr, c = sc;
    if (t < HEADS) {
        float a = lrelu(a_i[c * HEADS + t] + a_j[r * HEADS + t]);
        coef[t] = __expf(a - segmaxf[c * HEADS + t]) / segsum[c * HEADS + t];
    }
    __syncthreads();
    const int h = t >> 6;
    atomicAdd(&out[c * HD + t], coef[h] * xt[r * HD + t]);
}

extern "C" void kernel_launch(void* const* d_in, const int* in_sizes, int n_in,
                              void* d_out, int out_size, void* d_ws, size_t ws_size,
                              hipStream_t stream) {
    const float* x        = (const float*)d_in[0];   // [N, L, 300]
    const int*   ei       = (const int*)  d_in[1];   // [2, E] (JAX x64 off -> int32)
    const float* lin_w    = (const float*)d_in[2];   // [300, 300]
    const float* lin_b    = (const float*)d_in[3];   // [300]
    const float* gat_w    = (const float*)d_in[4];   // [300, 256]
    const float* att      = (const float*)d_in[5];   // [1, 4, 128]
    const float* gat_bias = (const float*)d_in[6];   // [256]
    float* out = (float*)d_out;

    // -------- workspace carve-up (256B aligned) --------
    char* ws = (char*)d_ws;
    auto take = [&](size_t bytes) {
        char* p = ws;
        ws += (bytes + 255) & ~(size_t)255;
        return p;
    };
    float*    xm          = (float*)   take((size_t)NN * INC * 4);   // 60.0 MB
    float*    xt          = (float*)   take((size_t)NN * HD  * 4);   // 51.2 MB
    float*    W2          = (float*)   take((size_t)INC * HD * 4);   // 0.3 MB
    float*    b2          = (float*)   take((size_t)HD * 4);
    float*    a_i         = (float*)   take((size_t)NN * HEADS * 4);
    float*    a_j         = (float*)   take((size_t)NN * HEADS * 4);
    float*    selfAlpha   = (float*)   take((size_t)NN * HEADS * 4);
    unsigned* segmax_bits = (unsigned*)take((size_t)NN * HEADS * 4);
    float*    segmaxf     = (float*)   take((size_t)NN * HEADS * 4);
    float*    segsum      = (float*)   take((size_t)NN * HEADS * 4);

    // K1: mean over L
    {
        long long tot = (long long)NN * INC;
        int blocks = (int)((tot + 255) / 256);
        mean_l_kernel<<<blocks, 256, 0, stream>>>(x, xm);
    }
    // K2: fold lin_w/gat_w and lin_b/gat_w
    {
        int tot = INC * HD + HD;
        fuse_weights_kernel<<<(tot + 255) / 256, 256, 0, stream>>>(lin_w, lin_b, gat_w, W2, b2);
    }
    // K3: WMMA fp32 GEMM  xt = xm @ W2 + b2
    {
        int blocks = (TOTWAVES + 7) / 8;
        xt_gemm_wmma<<<blocks, 256, 0, stream>>>(xm, W2, b2, xt);
    }
    // K4: per-node attention scalars
    att_scalars_kernel<<<NN, 256, 0, stream>>>(xt, att, a_i, a_j);
    // K5: self-loop init
    self_init_kernel<<<(NN * HEADS + 255) / 256, 256, 0, stream>>>(a_i, a_j, selfAlpha, segmax_bits);
    // K6: segment max over edges
    edge_max_kernel<<<(EE * HEADS + 255) / 256, 256, 0, stream>>>(ei, a_i, a_j, segmax_bits);
    // K7: segment sum init
    segsum_init_kernel<<<(NN * HEADS + 255) / 256, 256, 0, stream>>>(segmax_bits, selfAlpha, segmaxf, segsum);
    // K8: segment sum over edges
    edge_sum_kernel<<<(EE * HEADS + 255) / 256, 256, 0, stream>>>(ei, a_i, a_j, segmaxf, segsum);
    // K9: initialize output with self-loop contribution + bias
    {
        long long tot = (long long)NN * HD;
        out_init_kernel<<<(int)((tot + 255) / 256), 256, 0, stream>>>(selfAlpha, segmaxf, segsum, xt, gat_bias, out);
    }
    // K10: edge aggregation
    edge_aggr_kernel<<<EE, 256, 0, stream>>>(ei, a_i, a_j, segmaxf, segsum, xt, out);
}